// FineGrained2GNN_21852793602184
// MI455X (gfx1250) — compile-verified
//
#include <hip/hip_runtime.h>
#include <math.h>

// Problem constants (from reference)
#define NB      4096   // batches
#define NN      62     // graph nodes
#define NF      5      // input features
#define KF      20     // (K+1)*NF  (einsum GEMM K dim)
#define CO      64     // C_OUT     (einsum GEMM N dim)
#define NSEL    31
#define LSTR    64     // padded Laplacian stride (zero border rows/cols 62,63)
#define XSTR    21     // padded row stride for Chebyshev stacks
#define THREADS 256    // 8 wave32s

typedef float v2f __attribute__((ext_vector_type(2)));
typedef float v8f __attribute__((ext_vector_type(8)));

__device__ __forceinline__ v8f wmma4(v2f a, v2f b, v8f c) {
    // D = A(16x4,f32) * B(4x16,f32) + C(16x16,f32)   -- full fp32 accuracy
    return __builtin_amdgcn_wmma_f32_16x16x4_f32(false, a, false, b, (short)0, c, false, false);
}

// Chebyshev recursion on the matrix pipe.  Waves 0..3 each own one 16-row
// M-tile of:  x_k = (k==1 ? L@x0 : 2*L@x_{k-1} - x_{k-2}).
// sL:  64x64 padded Laplacian (A operand), sB0/sB1: 64x16 ping-pong B operand,
// sXA: einsum A-matrix [n][k*5+f] (also supplies the x_{k-2} term).
__device__ __forceinline__ void cheb_run(const float* sL, float* sB0, float* sB1,
                                         float* sXA, int wid, int half, int l16)
{
    float* bufs[2] = { sB0, sB1 };
    #pragma unroll
    for (int k = 1; k <= 3; ++k) {
        const int cur = (k + 1) & 1;          // k=1 reads buf0, k=2 buf1, k=3 buf0
        if (wid < 4) {
            const int   mt = wid;
            const float* Bc = bufs[cur];
            float*       Bn = bufs[cur ^ 1];
            v8f acc = {0.f,0.f,0.f,0.f,0.f,0.f,0.f,0.f};
            #pragma unroll
            for (int kc = 0; kc < 16; ++kc) {         // K = 64 (rows 62,63 zero)
                const int kb = kc * 4 + half * 2;     // lanes16-31 carry K2/K3
                v2f av, bv;
                av.x = sL[(mt * 16 + l16) * LSTR + kb];
                av.y = sL[(mt * 16 + l16) * LSTR + kb + 1];
                bv.x = Bc[kb * 16 + l16];
                bv.y = Bc[(kb + 1) * 16 + l16];
                acc = wmma4(av, bv, acc);
            }
            #pragma unroll
            for (int v = 0; v < 8; ++v) {             // D: M = v + 8*half, N = l16
                const int n = mt * 16 + v + 8 * half;
                if (l16 < NF && n < NN) {
                    float val = acc[v];
                    if (k >= 2) val = 2.0f * val - sXA[n * XSTR + (k - 2) * NF + l16];
                    Bn[n * 16 + l16]             = val;   // next-step B operand
                    sXA[n * XSTR + k * NF + l16] = val;   // einsum A-matrix
                }
            }
        }
        __syncthreads();
    }
}

__global__ __launch_bounds__(THREADS)
void cheb_moe_kernel(const float* __restrict__ x,
                     const float* __restrict__ adj,
                     const int*   __restrict__ mask_idx,
                     const float* __restrict__ gc_w,  const float* __restrict__ gc_b,
                     const float* __restrict__ gc1_w, const float* __restrict__ gc1_b,
                     const float* __restrict__ gc2_w, const float* __restrict__ gc2_b,
                     const float* __restrict__ fc1_w, const float* __restrict__ fc1_b,
                     const float* __restrict__ fc2_w, const float* __restrict__ fc2_b,
                     const float* __restrict__ cls_w, const float* __restrict__ cls_b,
                     float* __restrict__ out)
{
    __shared__ float sL[64 * LSTR];      // 16 KB  padded Laplacian (L1, then L2_b)
    __shared__ float sB0[64 * 16];       //  4 KB  Chebyshev B operand (ping)
    __shared__ float sB1[64 * 16];       //  4 KB  Chebyshev B operand (pong)
    __shared__ float sXA1[64 * XSTR];    //  5.25 KB  einsum A, graph1
    __shared__ float sXA2[64 * XSTR];    //  5.25 KB  einsum A, graph2 (masked)
    __shared__ float sW[3 * KF * CO];    // 15 KB  expert weights [kf][o]
    __shared__ float sM[64];             // node mask
    __shared__ float sDis[64];           // 1/sqrt(degree)
    __shared__ float sPart[8 * 8];       // per-wave partial logits
    __shared__ float sFinal[8];

    const int tid  = threadIdx.x;
    const int b    = blockIdx.x;
    const int lane = tid & 31;
    const int wid  = tid >> 5;
    const int half = lane >> 4;
    const int l16  = lane & 15;

    __builtin_prefetch(fc1_w, 0, 1);     // keep FC weights hot in L2
    __builtin_prefetch(fc2_w, 0, 1);
    __builtin_prefetch(cls_w, 0, 1);

    // ---- phase 1: zero-init padded buffers, stage expert weights
    for (int i = tid; i < 64 * LSTR; i += THREADS) sL[i] = 0.0f;
    for (int i = tid; i < 64 * 16;  i += THREADS) { sB0[i] = 0.0f; sB1[i] = 0.0f; }
    for (int i = tid; i < 64 * XSTR; i += THREADS) { sXA1[i] = 0.0f; sXA2[i] = 0.0f; }
    for (int i = tid; i < KF * CO; i += THREADS) {
        sW[0 * KF * CO + i] = gc1_w[i];   // e0 = expert1
        sW[1 * KF * CO + i] = gc2_w[i];   // e1 = expert2
        sW[2 * KF * CO + i] = gc_w[i];    // e2 = gate
    }
    if (tid < 64) sM[tid] = 0.0f;
    __syncthreads();

    // ---- phase 2: mask + unmasked degrees
    if (tid < NSEL) sM[mask_idx[b * NSEL + tid]] = 1.0f;   // duplicates benign
    if (tid < NN) {
        float d = 0.0f;
        for (int j = 0; j < NN; ++j) d += adj[tid * NN + j];
        sDis[tid] = (d > 0.0f) ? rsqrtf(d) : 0.0f;
    }
    __syncthreads();

    // ---- phase 3: sL <- L1; load x0 (and masked x0)
    for (int i = tid; i < NN * NN; i += THREADS) {
        int r = i / NN, c = i % NN;
        sL[r * LSTR + c] = -sDis[r] * sDis[c] * adj[i];
    }
    for (int t = tid; t < NN * NF; t += THREADS) {
        int n = t / NF, f = t % NF;
        float xv = x[(size_t)b * NN * NF + n * NF + f];
        sXA1[n * XSTR + f] = xv;
        sXA2[n * XSTR + f] = xv * sM[n];
        sB0[n * 16 + f]    = xv;
    }
    __syncthreads();

    // ---- phase 4: Chebyshev recursion, graph1 (WMMA)
    cheb_run(sL, sB0, sB1, sXA1, wid, half, l16);

    // ---- phase 5: masked degrees for L2_b
    if (tid < NN) {
        float d = 0.0f;
        for (int j = 0; j < NN; ++j) d += adj[tid * NN + j] * sM[j];
        d *= sM[tid];
        sDis[tid] = (d > 0.0f) ? rsqrtf(d) : 0.0f;
    }
    __syncthreads();

    // ---- phase 6: sL <- L2_b; reset B ping buffer with masked x0
    for (int i = tid; i < NN * NN; i += THREADS) {
        int r = i / NN, c = i % NN;
        sL[r * LSTR + c] = -sDis[r] * sM[r] * adj[i] * sM[c] * sDis[c];
    }
    for (int t = tid; t < NN * NF; t += THREADS) {
        int n = t / NF, f = t % NF;
        sB0[n * 16 + f] = sXA2[n * XSTR + f];
    }
    __syncthreads();

    // ---- phase 7: Chebyshev recursion, graph2 (WMMA)
    cheb_run(sL, sB0, sB1, sXA2, wid, half, l16);

    // ---- phase 8: einsum [62x20]@[20x64] per expert (WMMA) + fused FC epilogue.
    // Compile-time expert loop => static accumulator indexing, straight-line FMAs.
    float pl1[3] = {0.f, 0.f, 0.f};
    float pl2[3] = {0.f, 0.f, 0.f};
    float pg[2]  = {0.f, 0.f};
    #pragma unroll
    for (int e = 0; e < 3; ++e) {
        const float* A    = (e == 1) ? sXA2 : sXA1;
        const float* W    = sW + e * KF * CO;
        const float* bias = (e == 0) ? gc1_b : (e == 1) ? gc2_b : gc_b;
        const float* fw   = (e == 0) ? fc1_w : (e == 1) ? fc2_w : cls_w;
        #pragma unroll
        for (int tt = 0; tt < 2; ++tt) {          // 16 tiles / 8 waves = 2 each
            const int r  = wid * 2 + tt;
            const int mt = r >> 2;
            const int nt = r & 3;
            v8f acc = {0.f,0.f,0.f,0.f,0.f,0.f,0.f,0.f};
            #pragma unroll
            for (int kc = 0; kc < 5; ++kc) {      // K = 20 = 5 x 4
                const int kb = kc * 4 + half * 2;
                v2f av, bv;
                av.x = A[(mt * 16 + l16) * XSTR + kb];
                av.y = A[(mt * 16 + l16) * XSTR + kb + 1];
                bv.x = W[kb * CO + nt * 16 + l16];
                bv.y = W[(kb + 1) * CO + nt * 16 + l16];
                acc = wmma4(av, bv, acc);
            }
            const int   o  = nt * 16 + l16;
            const float bo = bias[o];
            #pragma unroll
            for (int v = 0; v < 8; ++v) {
                const int n = mt * 16 + v + 8 * half;
                if (n < NN) {
                    const float h = fmaxf(acc[v] + bo, 0.0f);
                    if (e == 0) {
                        const int idx = (n * CO + o) * 3;
                        pl1[0] += h * fw[idx];
                        pl1[1] += h * fw[idx + 1];
                        pl1[2] += h * fw[idx + 2];
                    } else if (e == 1) {
                        const int idx = (n * CO + o) * 3;
                        pl2[0] += h * fw[idx];
                        pl2[1] += h * fw[idx + 1];
                        pl2[2] += h * fw[idx + 2];
                    } else {
                        const int idx = (n * CO + o) * 2;
                        pg[0] += h * fw[idx];
                        pg[1] += h * fw[idx + 1];
                    }
                }
            }
        }
    }

    // ---- phase 9: wave32 butterfly reduction, then tiny cross-wave sum
    float vals[8] = { pl1[0], pl1[1], pl1[2], pl2[0], pl2[1], pl2[2], pg[0], pg[1] };
    #pragma unroll
    for (int i = 0; i < 8; ++i) {
        float v = vals[i];
        #pragma unroll
        for (int off = 16; off > 0; off >>= 1) v += __shfl_xor(v, off, 32);
        if (lane == 0) sPart[wid * 8 + i] = v;
    }
    __syncthreads();
    if (tid < 8) {
        float s = 0.0f;
        for (int w = 0; w < 8; ++w) s += sPart[w * 8 + tid];
        sFinal[tid] = s;
    }
    __syncthreads();

    // ---- phase 10: softmax gate + mixture, write 3 outputs
    if (tid == 0) {
        float l1v[3], l2v[3];
        #pragma unroll
        for (int c = 0; c < 3; ++c) {
            l1v[c] = sFinal[c]     + fc1_b[c];
            l2v[c] = sFinal[3 + c] + fc2_b[c];
        }
        float g0 = sFinal[6] + cls_b[0];
        float g1 = sFinal[7] + cls_b[1];
        float mx  = fmaxf(g0, g1);
        float e0  = __expf(g0 - mx), e1 = __expf(g1 - mx);
        float inv = 1.0f / (e0 + e1);
        float p0  = e0 * inv, p1 = e1 * inv;
        #pragma unroll
        for (int c = 0; c < 3; ++c) out[b * 3 + c] = l1v[c] * p0 + l2v[c] * p1;
    }
}

extern "C" void kernel_launch(void* const* d_in, const int* in_sizes, int n_in,
                              void* d_out, int out_size, void* d_ws, size_t ws_size,
                              hipStream_t stream) {
    const float* x        = (const float*)d_in[0];
    const float* adj      = (const float*)d_in[1];
    const int*   mask_idx = (const int*)  d_in[2];
    const float* gc_w     = (const float*)d_in[3];
    const float* gc_b     = (const float*)d_in[4];
    const float* gc1_w    = (const float*)d_in[5];
    const float* gc1_b    = (const float*)d_in[6];
    const float* gc2_w    = (const float*)d_in[7];
    const float* gc2_b    = (const float*)d_in[8];
    const float* fc1_w    = (const float*)d_in[9];
    const float* fc1_b    = (const float*)d_in[10];
    const float* fc2_w    = (const float*)d_in[11];
    const float* fc2_b    = (const float*)d_in[12];
    const float* cls_w    = (const float*)d_in[13];
    const float* cls_b    = (const float*)d_in[14];
    float* out = (float*)d_out;

    cheb_moe_kernel<<<NB, THREADS, 0, stream>>>(
        x, adj, mask_idx, gc_w, gc_b, gc1_w, gc1_b, gc2_w, gc2_b,
        fc1_w, fc1_b, fc2_w, fc2_b, cls_w, cls_b, out);
}